// Potential_15874199126158
// MI455X (gfx1250) — compile-verified
//
#include <hip/hip_runtime.h>

// CDNA5 / gfx1250: wave32, WMMA 16x16x4 f32.
typedef float v2f __attribute__((ext_vector_type(2)));
typedef float v8f __attribute__((ext_vector_type(8)));

#define WAVES_PER_BLOCK 8
#define BLOCK (WAVES_PER_BLOCK * 32)
#define M_MAX 2048   // reference: 2048 anchors; 3*2048*8 = 48KB LDS

// out[n] = sum_m w_m * exp(-0.5*||x_n - y_m||^2 / sigma_m^2)
// d2 tile via V_WMMA_F32_16X16X4_F32 with the K=4 embedding:
//   A[n][.] = [x0, x1, x2, 1],  B[.][m] = [-2y0, -2y1, -2y2, |y|^2],  C = row norms
__global__ __launch_bounds__(BLOCK)
void potential_rbf_wmma(const float* __restrict__ loc,    // [N,3]
                        const float* __restrict__ aloc,   // [M,3]
                        const float* __restrict__ aw,     // [M]
                        const float* __restrict__ awid,   // [M]
                        float* __restrict__ out,          // [N]
                        int N, int M)
{
    // Packed per-anchor constants (padded to a multiple of 16 with zeros so the
    // hot loop needs no bounds checks and EXEC never diverges).
    __shared__ v2f sBlo[M_MAX];   // (-2*y0, -2*y1)   -> B rows K=0,1 (lanes 0-15)
    __shared__ v2f sBhi[M_MAX];   // (-2*y2, |y|^2)   -> B rows K=2,3 (lanes 16-31)
    __shared__ v2f sCW[M_MAX];    // (-0.5*log2e/w^2, weight)

    const int Mpad = (M + 15) & ~15;

    for (int m = threadIdx.x; m < Mpad; m += BLOCK) {
        v2f blo = {0.f, 0.f}, bhi = {0.f, 0.f}, cw = {0.f, 0.f};
        if (m < M) {
            float y0 = aloc[3 * m + 0];
            float y1 = aloc[3 * m + 1];
            float y2 = aloc[3 * m + 2];
            blo.x = -2.0f * y0;
            blo.y = -2.0f * y1;
            bhi.x = -2.0f * y2;
            bhi.y = y0 * y0 + y1 * y1 + y2 * y2;
            float w = awid[m];
            cw.x = -0.72134752044448170f / (w * w);  // -0.5*log2(e)/w^2
            cw.y = aw[m];
        }
        sBlo[m] = blo;
        sBhi[m] = bhi;
        sCW[m]  = cw;
    }
    __syncthreads();

    const int lane = threadIdx.x & 31;
    const int wave = threadIdx.x >> 5;
    const int rowBase = (blockIdx.x * WAVES_PER_BLOCK + wave) * 16;

    if (rowBase < N) {                 // wave-uniform guard: EXEC stays all-ones inside
        const int col = lane & 15;     // column index for B / C / D
        const int hi  = lane >> 4;     // lane-group select (K=0,1 vs K=2,3; row+0 vs row+8)

        // A matrix: 16x4 f32. Lane holds row rowBase+col.
        int r = rowBase + col;
        int rc = r < (N - 1) ? r : (N - 1);  // clamp tail reads (writes guarded below)
        float x0 = loc[3 * rc + 0];
        float x1 = loc[3 * rc + 1];
        float x2 = loc[3 * rc + 2];
        float nrm = x0 * x0 + x1 * x1 + x2 * x2;

        v2f A;
        A.x = hi ? x2 : x0;            // V0: K=0 (lanes 0-15) / K=2 (lanes 16-31)
        A.y = hi ? 1.0f : x1;          // V1: K=1               / K=3

        // C matrix: C[v] = |x_row|^2 where row = rowBase + v + 8*hi.
        // Row r's norm lives in lane (r - rowBase) -> gather via ds_bpermute.
        v8f C;
#pragma unroll
        for (int v = 0; v < 8; ++v)
            C[v] = __shfl(nrm, v + 8 * hi, 32);

        // Per-lane B base pointer chosen once (no in-loop selects).
        const v2f* __restrict__ pB = hi ? sBhi : sBlo;

        v8f acc = {0.f, 0.f, 0.f, 0.f, 0.f, 0.f, 0.f, 0.f};

        for (int mb = 0; mb < Mpad; mb += 16) {
            int m = mb + col;
            v2f Bv = pB[m];            // one ds_load_b64
            // D = A*B + C  => full 16x16 tile of squared distances.
            v8f D = __builtin_amdgcn_wmma_f32_16x16x4_f32(
                false, A, false, Bv, (short)0, C, false, false);
            v2f cw = sCW[m];           // one ds_load_b64: (scale, weight)
#pragma unroll
            for (int v = 0; v < 8; ++v)
                acc[v] += cw.y * __builtin_amdgcn_exp2f(D[v] * cw.x);
        }

        // Reduce the 16 columns held across each 16-lane group (masks 1..8 stay in-group).
#pragma unroll
        for (int v = 0; v < 8; ++v) {
            float s = acc[v];
            s += __shfl_xor(s, 8, 32);
            s += __shfl_xor(s, 4, 32);
            s += __shfl_xor(s, 2, 32);
            s += __shfl_xor(s, 1, 32);
            int row = rowBase + 8 * hi + v;
            if (col == 0 && row < N)
                out[row] = s;
        }
    }
}

extern "C" void kernel_launch(void* const* d_in, const int* in_sizes, int n_in,
                              void* d_out, int out_size, void* d_ws, size_t ws_size,
                              hipStream_t stream)
{
    const float* loc  = (const float*)d_in[0];   // [N,3]
    const float* aloc = (const float*)d_in[1];   // [M,3]
    const float* aw   = (const float*)d_in[2];   // [M]
    const float* awid = (const float*)d_in[3];   // [M]
    float* out = (float*)d_out;

    int N = in_sizes[0] / 3;
    int M = in_sizes[2];

    int rowsPerBlock = WAVES_PER_BLOCK * 16;     // 128 rows per block
    int grid = (N + rowsPerBlock - 1) / rowsPerBlock;
    potential_rbf_wmma<<<grid, BLOCK, 0, stream>>>(loc, aloc, aw, awid, out, N, M);
}